// MaskedCrossEntropy_50354196578629
// MI455X (gfx1250) — compile-verified
//
#include <hip/hip_runtime.h>
#include <hip/hip_bf16.h>

// MaskedCrossEntropy for MI455X (gfx1250, wave32).
// Stage 1: one wave32 per row, single pass, online softmax + masked dot sums.
//          128-bit non-temporal loads (data touched once; 524MB >> 192MB L2),
//          single v_exp_f32 per element, uniform trip count + unroll for MLW.
// Stage 2: one wave sums 8192 per-row losses with V_WMMA_F32_16X16X4_F32
//          (B = ones => D accumulates row-sums), then writes mean.

typedef __attribute__((ext_vector_type(2))) float v2f;
typedef __attribute__((ext_vector_type(4))) float v4f;
typedef __attribute__((ext_vector_type(8))) float v8f;

#define LOG2E  1.4426950408889634f
#define LN2    0.6931471805599453f
#define NEGBIG (-3.402823466e38f)
#define MASK_EPS (-1.0e-6f)

// Online-softmax accumulate with exactly ONE exp2 per element.
// Invariant: s = sum_mask exp(x_i - m), m = max over masked so far (or NEGBIG).
__device__ __forceinline__ void mce_accum(float x, float t,
                                          float& m, float& s, float& T, float& U) {
    bool  msk = (t >= MASK_EPS);
    float xm  = msk ? x : NEGBIG;          // unmasked => xm <= m always
    float tm  = msk ? t : 0.0f;
    float m2  = fmaxf(m, xm);
    float lo  = fminf(m, xm);
    float d   = __builtin_amdgcn_exp2f((lo - m2) * LOG2E);  // exp(min-max)
    bool  ge  = (xm > m);                  // new max (implies msk)
    // ge:  s' = s*d + 1      (rescale old sum, new elem contributes 1)
    // !ge: s' = s + msk?d:0  (d = exp(xm - m))
    s = ge ? fmaf(s, d, 1.0f) : (s + (msk ? d : 0.0f));
    m = m2;
    T += tm;
    U = fmaf(tm, x, U);
}

__global__ __launch_bounds__(256)
void mce_row_loss_kernel(const float* __restrict__ X, const float* __restrict__ Tg,
                         float* __restrict__ row_loss, int N, int C) {
    const int lane   = threadIdx.x & 31;
    const int waveId = (int)((blockIdx.x * blockDim.x + threadIdx.x) >> 5);
    const int nWaves = (int)((gridDim.x * blockDim.x) >> 5);
    const int nv     = C >> 2;      // float4 groups per row (C % 4 == 0)
    const int full   = nv >> 5;     // uniform iterations (32 float4 per wave-iter)

    for (int row = waveId; row < N; row += nWaves) {
        const v4f* xr = (const v4f*)(X  + (size_t)row * (size_t)C);
        const v4f* tr = (const v4f*)(Tg + (size_t)row * (size_t)C);

        float m = NEGBIG, s = 0.0f, T = 0.0f, U = 0.0f;

        #pragma unroll 2
        for (int i = 0; i < full; ++i) {
            int v = (i << 5) + lane;
            v4f x4 = __builtin_nontemporal_load(xr + v);
            v4f t4 = __builtin_nontemporal_load(tr + v);
            mce_accum(x4.x, t4.x, m, s, T, U);
            mce_accum(x4.y, t4.y, m, s, T, U);
            mce_accum(x4.z, t4.z, m, s, T, U);
            mce_accum(x4.w, t4.w, m, s, T, U);
        }
        // predicated tail: remaining nv - full*32 float4 groups
        {
            int v = (full << 5) + lane;
            if (v < nv) {
                v4f x4 = __builtin_nontemporal_load(xr + v);
                v4f t4 = __builtin_nontemporal_load(tr + v);
                mce_accum(x4.x, t4.x, m, s, T, U);
                mce_accum(x4.y, t4.y, m, s, T, U);
                mce_accum(x4.z, t4.z, m, s, T, U);
                mce_accum(x4.w, t4.w, m, s, T, U);
            }
        }

        // wave32 butterfly reduction: (m,s) softmax-merge; T,U plain sums
        #pragma unroll
        for (int off = 16; off; off >>= 1) {
            float mo = __shfl_xor(m, off, 32);
            float so = __shfl_xor(s, off, 32);
            float m2 = fmaxf(m, mo);
            s = s  * __builtin_amdgcn_exp2f((m  - m2) * LOG2E)
              + so * __builtin_amdgcn_exp2f((mo - m2) * LOG2E);
            m = m2;
            T += __shfl_xor(T, off, 32);
            U += __shfl_xor(U, off, 32);
        }

        if (lane == 0) {
            // lse = m + ln(s);  per_row = lse*T - U  (0 if row fully masked)
            float lse  = m + __builtin_amdgcn_logf(s) * LN2;
            float loss = (s > 0.0f) ? fmaf(lse, T, -U) : 0.0f;
            row_loss[row] = loss;
        }
    }
}

// Single-wave reduction of N per-row losses -> mean.
// Chunks of 64: A(16x4) per-lane layout: lanes 0-15 hold row m=lane, K=0..1;
// lanes 16-31 hold row m=lane-16, K=2..3. B = ones => D[m][n] = rowsum(A)[m],
// accumulated across chunks in C. Sum of all 256 C entries = 16 * total.
__global__ __launch_bounds__(32)
void mce_reduce_mean_kernel(const float* __restrict__ row_loss,
                            float* __restrict__ out, int N) {
    const int lane   = threadIdx.x & 31;
    const int m      = lane & 15;
    const int kbase  = (lane >> 4) << 1;   // 0 or 2
    const int chunks = N >> 6;             // 64 values per WMMA chunk
    float total;

#if __has_builtin(__builtin_amdgcn_wmma_f32_16x16x4_f32)
    v8f c = {};
    v2f b; b[0] = 1.0f; b[1] = 1.0f;
    for (int ch = 0; ch < chunks; ++ch) {
        const float* p = row_loss + ch * 64 + m * 4 + kbase;
        v2f a; a[0] = p[0]; a[1] = p[1];
        c = __builtin_amdgcn_wmma_f32_16x16x4_f32(
                /*neg_a=*/false, a, /*neg_b=*/false, b,
                /*c_mod=*/(short)0, c, /*reuse_a=*/false, /*reuse_b=*/false);
    }
    float part = c[0] + c[1] + c[2] + c[3] + c[4] + c[5] + c[6] + c[7];
    #pragma unroll
    for (int off = 16; off; off >>= 1) part += __shfl_xor(part, off, 32);
    total = part * (1.0f / 16.0f);
#else
    float part = 0.0f;
    for (int i = lane; i < (chunks << 6); i += 32) part += row_loss[i];
    #pragma unroll
    for (int off = 16; off; off >>= 1) part += __shfl_xor(part, off, 32);
    total = part;
#endif

    // tail (N % 64), plain sum
    float tail = 0.0f;
    for (int i = (chunks << 6) + lane; i < N; i += 32) tail += row_loss[i];
    #pragma unroll
    for (int off = 16; off; off >>= 1) tail += __shfl_xor(tail, off, 32);

    if (lane == 0) out[0] = (total + tail) / (float)N;
}

extern "C" void kernel_launch(void* const* d_in, const int* in_sizes, int n_in,
                              void* d_out, int out_size, void* d_ws, size_t ws_size,
                              hipStream_t stream) {
    const float* logits  = (const float*)d_in[0];
    const float* targets = (const float*)d_in[1];
    float* out = (float*)d_out;
    float* ws  = (float*)d_ws;   // N floats of per-row loss

    const int C = 8000;                       // reference: [N, 8000]
    const int N = in_sizes[0] / C;            // N = 8192

    int blocks = (N + 7) / 8;                 // 8 waves/block, one row per wave
    mce_row_loss_kernel<<<blocks, 256, 0, stream>>>(logits, targets, ws, N, C);
    mce_reduce_mean_kernel<<<1, 32, 0, stream>>>(ws, out, N);
}